// Encoder_52106543235873
// MI455X (gfx1250) — compile-verified
//
#include <hip/hip_runtime.h>
#include <math.h>

typedef float v2f __attribute__((ext_vector_type(2)));
typedef float v8f __attribute__((ext_vector_type(8)));

namespace {
constexpr int kB  = 64;    // batch
constexpr int kT  = 512;   // timesteps
constexpr int kF  = 512;   // input features
constexpr int kU  = 1024;  // hidden units
constexpr int kG3 = 3 * kU;       // 3072 gate columns
constexpr int kUT = 16;           // hidden columns owned per workgroup
constexpr int kNWG = kU / kUT;    // 64 persistent workgroups
constexpr int kThreads = 384;     // 12 waves = 4 batch tiles x 3 gates
constexpr int kCols = 3 * kUT;    // 48 gate-columns per WG (z,r,h)

// LDS layouts: column-major arrays of K-pairs (float2 = rows 2p,2p+1 of one col)
// stride padded so per-lane column stride (in dwords) % 64 == 4 -> spread banks
constexpr int kWHPairs  = kU / 2;         // 512
constexpr int kWHStride = kWHPairs + 2;   // 514
constexpr int kWIPairs  = kF / 2;         // 256
constexpr int kWIStride = kWIPairs + 2;   // 258
constexpr size_t kSmemBytes =
    sizeof(float2) * (size_t)kCols * kWHStride +   // 197376 B  (w_h slice)
    sizeof(float2) * (size_t)kCols * kWIStride +   //  99072 B  (w_i slice)
    sizeof(float)  * 2 * 3 * kB * kUT;             //  24576 B  (rec/xs staging)
}

__device__ __forceinline__ v2f mk2(float a, float b) { v2f r; r.x = a; r.y = b; return r; }

// Device-wide barrier for the persistent cooperative kernel.
// release fence -> global_wb (make h_next visible at DEV scope)
// acquire fence -> global_inv (drop stale h lines from this WGP's caches)
__device__ __forceinline__ void grid_barrier(int* cnt, volatile int* gen, int nwg) {
  __builtin_amdgcn_fence(__ATOMIC_RELEASE, "agent");
  __syncthreads();
  if (threadIdx.x == 0) {
    int g = *gen;
    if (atomicAdd(cnt, 1) == nwg - 1) {
      *cnt = 0;
      __builtin_amdgcn_fence(__ATOMIC_RELEASE, "agent");
      atomicExch((int*)gen, g + 1);
    } else {
      while (*gen == g) { __builtin_amdgcn_s_sleep(2); }
    }
  }
  __syncthreads();
  __builtin_amdgcn_fence(__ATOMIC_ACQUIRE, "agent");
}

__global__ void gru_init(const float* __restrict__ hidden,
                         float* __restrict__ hbuf0, int* __restrict__ bar) {
  int i = blockIdx.x * blockDim.x + threadIdx.x;
  if (i < kB * kU) hbuf0[i] = hidden[i];
  if (i == 0) { bar[0] = 0; bar[1] = 0; }
}

__global__ __launch_bounds__(kThreads, 1)
void gru_scan_kernel(const float* __restrict__ x,      // [B,T,F]
                     const float* __restrict__ w_i,    // [F,3U]
                     const float* __restrict__ w_h,    // [U,3U]
                     const float* __restrict__ b_i,    // [3U]
                     const float* __restrict__ b_h,    // [3U]
                     const float* __restrict__ mask,   // [B,T,U]
                     float* __restrict__ out,          // [B,T,U]
                     float* __restrict__ state_out,    // [B,U]
                     float* __restrict__ hbuf0,
                     float* __restrict__ hbuf1,
                     int* bar_cnt, volatile int* bar_gen) {
  extern __shared__ __align__(16) char smem_raw[];
  float2* whp   = (float2*)smem_raw;                       // [48][514]
  float2* wip   = whp + kCols * kWHStride;                 // [48][258]
  float*  rec_s = (float*)(wip + kCols * kWIStride);       // [3][64][16]
  float*  xs_s  = rec_s + 3 * kB * kUT;                    // [3][64][16]

  const int tid = threadIdx.x;
  const int u0  = blockIdx.x * kUT;

  // ---- one-time: stage this WG's w_h / w_i column slices into LDS, K-paired,
  // column-major so GEMM B-fragments are single 16B ds loads ----
  for (int i = tid; i < kCols * kWHPairs; i += kThreads) {
    int c = i % kCols, p = i / kCols;
    int gc = (c >> 4) * kU + u0 + (c & 15);
    whp[c * kWHStride + p] = make_float2(w_h[(size_t)(2 * p) * kG3 + gc],
                                         w_h[(size_t)(2 * p + 1) * kG3 + gc]);
  }
  for (int i = tid; i < kCols * kWIPairs; i += kThreads) {
    int c = i % kCols, p = i / kCols;
    int gc = (c >> 4) * kU + u0 + (c & 15);
    wip[c * kWIStride + p] = make_float2(w_i[(size_t)(2 * p) * kG3 + gc],
                                         w_i[(size_t)(2 * p + 1) * kG3 + gc]);
  }
  __syncthreads();

  const int lane = tid & 31;
  const int wave = tid >> 5;   // 0..11
  const int mt   = wave & 3;   // batch tile (16 rows)
  const int gate = wave >> 2;  // 0=z 1=r 2=h
  const int rc   = lane & 15;  // A-row within tile / B-column within tile
  const int chi  = lane >> 4;  // K half-selector

  const float2* whcol = whp + (gate * kUT + rc) * kWHStride;
  const float2* wicol = wip + (gate * kUT + rc) * kWIStride;
  const float*  xrow0 = x + (size_t)(mt * 16 + rc) * kT * kF + 4 * chi;

  for (int t = 0; t < kT; ++t) {
    const float* hcur = (t & 1) ? hbuf1 : hbuf0;
    float*       hnxt = (t & 1) ? hbuf0 : hbuf1;
    const float* xrow = xrow0 + (size_t)t * kF;

    if (t + 1 < kT) __builtin_prefetch(xrow + kF, 0, 1);  // global_prefetch_b8

    // --- rec tile: 16x16 = h[mt] (16x1024) @ w_h[:, gate cols] ---
    // K relabeling per 8: chi=0 owns K=8s..8s+3, chi=1 owns K=8s+4..8s+7;
    // WMMA#1 eats .xy of each float4, WMMA#2 eats .zw. A and B agree.
    v8f accR = {0.f, 0.f, 0.f, 0.f, 0.f, 0.f, 0.f, 0.f};
    const float* arow = hcur + (size_t)(mt * 16 + rc) * kU + 4 * chi;
#pragma unroll 4
    for (int s8 = 0; s8 < kU; s8 += 8) {
      float4 a = *(const float4*)(arow + s8);
      float4 b = *(const float4*)(whcol + (s8 >> 1) + 2 * chi);
      accR = __builtin_amdgcn_wmma_f32_16x16x4_f32(
          false, mk2(a.x, a.y), false, mk2(b.x, b.y), (short)0, accR, false, false);
      accR = __builtin_amdgcn_wmma_f32_16x16x4_f32(
          false, mk2(a.z, a.w), false, mk2(b.z, b.w), (short)0, accR, false, false);
    }

    // --- xs tile: 16x16 = x[mt, t] (16x512) @ w_i[:, gate cols] ---
    v8f accX = {0.f, 0.f, 0.f, 0.f, 0.f, 0.f, 0.f, 0.f};
#pragma unroll 4
    for (int s8 = 0; s8 < kF; s8 += 8) {
      float4 a = *(const float4*)(xrow + s8);
      float4 b = *(const float4*)(wicol + (s8 >> 1) + 2 * chi);
      accX = __builtin_amdgcn_wmma_f32_16x16x4_f32(
          false, mk2(a.x, a.y), false, mk2(b.x, b.y), (short)0, accX, false, false);
      accX = __builtin_amdgcn_wmma_f32_16x16x4_f32(
          false, mk2(a.z, a.w), false, mk2(b.z, b.w), (short)0, accX, false, false);
    }

    // --- stage tiles to LDS (C layout: VGPR v -> M=v (lanes 0-15) / v+8) ---
#pragma unroll
    for (int v = 0; v < 8; ++v) {
      int m = mt * 16 + v + 8 * chi;
      rec_s[gate * (kB * kUT) + m * kUT + rc] = accR[v];
      xs_s [gate * (kB * kUT) + m * kUT + rc] = accX[v];
    }
    __syncthreads();

    // --- elementwise GRU cell over this WG's 64x16 slice ---
    for (int e = tid; e < kB * kUT; e += kThreads) {
      int b = e >> 4, c = e & 15;
      int u = u0 + c;
      float xz = xs_s[e]                  + b_i[u];
      float xr = xs_s[kB * kUT + e]       + b_i[kU + u];
      float xh = xs_s[2 * kB * kUT + e]   + b_i[2 * kU + u];
      float rz = rec_s[e]                 + b_h[u];
      float rr = rec_s[kB * kUT + e]      + b_h[kU + u];
      float rh = rec_s[2 * kB * kUT + e]  + b_h[2 * kU + u];
      float z  = tanhf(xz + rz);
      float r  = tanhf(xr + rr);
      float hh = tanhf(xh + r * rh);
      float hp = hcur[b * kU + u];
      float hn = z * hp + (1.0f - z) * hh;
      hnxt[b * kU + u] = hn;
      size_t oi = ((size_t)b * kT + t) * kU + u;
      __builtin_nontemporal_store(hn * mask[oi], out + oi);
      if (t == kT - 1) state_out[b * kU + u] = hn;
    }

    if (t + 1 < kT) grid_barrier(bar_cnt, bar_gen, kNWG);
  }
}

extern "C" void kernel_launch(void* const* d_in, const int* in_sizes, int n_in,
                              void* d_out, int out_size, void* d_ws, size_t ws_size,
                              hipStream_t stream) {
  (void)in_sizes; (void)n_in; (void)out_size; (void)ws_size;
  const float* x      = (const float*)d_in[0];
  const float* hidden = (const float*)d_in[1];
  const float* w_i    = (const float*)d_in[2];
  const float* w_h    = (const float*)d_in[3];
  const float* b_i    = (const float*)d_in[4];
  const float* b_h    = (const float*)d_in[5];
  const float* mask   = (const float*)d_in[6];

  float* out   = (float*)d_out;
  float* state = out + (size_t)kB * kT * kU;

  char* ws = (char*)d_ws;
  int*   bar   = (int*)ws;                       // [cnt, gen]
  float* hbuf0 = (float*)(ws + 256);
  float* hbuf1 = hbuf0 + kB * kU;

  gru_init<<<(kB * kU + 255) / 256, 256, 0, stream>>>(hidden, hbuf0, bar);
  gru_scan_kernel<<<kNWG, kThreads, kSmemBytes, stream>>>(
      x, w_i, w_h, b_i, b_h, mask, out, state, hbuf0, hbuf1, bar, bar + 1);
}